// HyperMatrix_3917010174731
// MI455X (gfx1250) — compile-verified
//
#include <hip/hip_runtime.h>
#include <cstdint>

// ---------------- problem constants ----------------
#define B_DIM   32
#define IN_DIM  2048
#define OUT_DIM 2048
#define Z_DIMN  64
#define NNZ     209715
#define TK      256                      // nonzeros per workgroup tile
#define NTILES  ((NNZ + TK - 1) / TK)    // 820

typedef float        v2f   __attribute__((ext_vector_type(2)));
typedef float        v8f   __attribute__((ext_vector_type(8)));
typedef unsigned int u32x4 __attribute__((ext_vector_type(4)));
typedef int          i32x4 __attribute__((ext_vector_type(4)));
typedef int          i32x8 __attribute__((ext_vector_type(8)));

// ---------------- pre: transpose x -> xT [IN,32], zero outT [OUT,32] ----------------
__global__ __launch_bounds__(256) void hyper_pre(const float* __restrict__ x,
                                                 float* __restrict__ xT,
                                                 float* __restrict__ outT) {
    int i = blockIdx.x * 256 + threadIdx.x;          // 0 .. 65535
    int b = i / IN_DIM, c = i % IN_DIM;
    xT[c * B_DIM + b] = x[i];
    outT[i] = 0.0f;
}

// ---------------- main fused kernel ----------------
__global__ __launch_bounds__(256) void hyper_fused(const float* __restrict__ xT,
                                                   const float* __restrict__ z,
                                                   const float* __restrict__ Wnet,
                                                   const float* __restrict__ bnet,
                                                   const int*   __restrict__ in_idx,
                                                   const int*   __restrict__ out_idx,
                                                   float* __restrict__ outT) {
    __shared__ float zs [B_DIM * Z_DIMN];     //  8 KB : z[b][d]
    __shared__ float wsm[Z_DIMN * TK];        // 64 KB : W tile, [d][kk]
    __shared__ float es [TK * (B_DIM + 1)];   // 33 KB : E[kk][b] (+1 pad)

    const int tid  = threadIdx.x;
    const int wave = tid >> 5;
    const int lane = tid & 31;
    const int half = lane >> 4;
    const int l    = lane & 15;
    const int k0   = blockIdx.x * TK;

    // ---- stage z into LDS (cooperative, coalesced) ----
    for (int i = tid; i < B_DIM * Z_DIMN; i += 256) zs[i] = z[i];

    // ---- TDM: async 2D tile load W[0:64, k0:k0+TK] -> LDS (wave 0 issues) ----
    if (wave == 0) {
        unsigned long long gaddr = (unsigned long long)(const void*)(Wnet + k0);
        unsigned int ldsaddr = (unsigned int)(unsigned long long)(const void*)&wsm[0];
        u32x4 g0;
        g0.x = 1u;                                            // count=1, user mode
        g0.y = ldsaddr;                                       // LDS dest
        g0.z = (unsigned int)(gaddr & 0xFFFFFFFFull);         // global addr lo
        g0.w = (unsigned int)((gaddr >> 32) & 0x1FFFFFFull)   // global addr hi
             | (2u << 30);                                    // type = 2 ("image")
        i32x8 g1;
        g1[0] = (int)(2u << 16);                              // data_size = 4B
        g1[1] = (int)(((unsigned)NNZ & 0xFFFFu) << 16);       // tensor_dim0 lo16
        g1[2] = (int)(((unsigned)NNZ >> 16) |                 // tensor_dim0 hi16
                      ((unsigned)Z_DIMN << 16));              // tensor_dim1 lo16
        g1[3] = (int)((unsigned)TK << 16);                    // tile_dim0 = 256
        g1[4] = (int)Z_DIMN;                                  // tile_dim1 = 64 (dim2=0)
        g1[5] = (int)NNZ;                                     // tensor_dim0_stride lo32
        g1[6] = 0;                                            // stride hi16, dim1_stride lo
        g1[7] = 0;
        i32x4 gz = {0, 0, 0, 0};
#if __clang_major__ >= 23
        i32x8 gz8 = {0, 0, 0, 0, 0, 0, 0, 0};
        __builtin_amdgcn_tensor_load_to_lds(g0, g1, gz, gz, gz8, 0);
#else
        __builtin_amdgcn_tensor_load_to_lds(g0, g1, gz, gz, 0);
#endif
        __builtin_amdgcn_s_wait_tensorcnt(0);
    }
    __syncthreads();

    // ---- WMMA phase: E[32 x TK] = z @ Wtile, fp32 16x16x4, K = 64 = 16 steps ----
    // wave handles N-tiles {2w, 2w+1} x M-tiles {0,1}
    const int n0 = wave * 2;
    v8f acc[2][2] = {};
#pragma unroll
    for (int j = 0; j < 16; ++j) {
        const int kb = 4 * j + 2 * half;   // this lane-half's K pair
        // A (16x4 f32): lane l holds z[M=l][kb], z[M=l][kb+1] (contiguous -> b64)
        v2f a0 = *(const v2f*)&zs[(0 * 16 + l) * Z_DIMN + kb];
        v2f a1 = *(const v2f*)&zs[(1 * 16 + l) * Z_DIMN + kb];
        // B (4x16 f32): lane l holds W[kb][n], W[kb+1][n]
        v2f b0, b1;
        b0.x = wsm[(kb + 0) * TK + (n0 + 0) * 16 + l];
        b0.y = wsm[(kb + 1) * TK + (n0 + 0) * 16 + l];
        b1.x = wsm[(kb + 0) * TK + (n0 + 1) * 16 + l];
        b1.y = wsm[(kb + 1) * TK + (n0 + 1) * 16 + l];
        acc[0][0] = __builtin_amdgcn_wmma_f32_16x16x4_f32(false, a0, false, b0,
                                                          (short)0, acc[0][0], false, false);
        acc[1][0] = __builtin_amdgcn_wmma_f32_16x16x4_f32(false, a1, false, b0,
                                                          (short)0, acc[1][0], false, false);
        acc[0][1] = __builtin_amdgcn_wmma_f32_16x16x4_f32(false, a0, false, b1,
                                                          (short)0, acc[0][1], false, false);
        acc[1][1] = __builtin_amdgcn_wmma_f32_16x16x4_f32(false, a1, false, b1,
                                                          (short)0, acc[1][1], false, false);
    }

    // ---- bias + spill E to LDS as es[kk][batch] ----
    const int kk0 = (n0 + 0) * 16 + l;
    const int kk1 = (n0 + 1) * 16 + l;
    const float bias0 = (k0 + kk0 < NNZ) ? bnet[k0 + kk0] : 0.0f;
    const float bias1 = (k0 + kk1 < NNZ) ? bnet[k0 + kk1] : 0.0f;
#pragma unroll
    for (int r = 0; r < 8; ++r) {
        const int brow = r + 8 * half;     // batch row within an M-tile (D layout)
        es[kk0 * (B_DIM + 1) + 0  + brow] = acc[0][0][r] + bias0;
        es[kk0 * (B_DIM + 1) + 16 + brow] = acc[1][0][r] + bias0;
        es[kk1 * (B_DIM + 1) + 0  + brow] = acc[0][1][r] + bias1;
        es[kk1 * (B_DIM + 1) + 16 + brow] = acc[1][1][r] + bias1;
    }
    __syncthreads();

    // ---- scatter phase: lane = batch, coalesced 128B gather + fp32 atomic add ----
#pragma unroll 4
    for (int i = 0; i < TK / 8; ++i) {     // 32 nonzeros per wave
        const int kk = wave * (TK / 8) + i;
        const int k  = k0 + kk;
        if (k >= NNZ) break;
        const int ii = in_idx[k];
        const int oi = out_idx[k];
        const float e  = es[kk * (B_DIM + 1) + lane];
        const float xv = xT[ii * B_DIM + lane];
        unsafeAtomicAdd(&outT[oi * B_DIM + lane], e * xv);
    }
}

// ---------------- post: transpose outT [OUT,32] -> out [32,OUT] ----------------
__global__ __launch_bounds__(256) void hyper_post(const float* __restrict__ outT,
                                                  float* __restrict__ out) {
    int i = blockIdx.x * 256 + threadIdx.x;          // i = b*OUT + o
    int b = i / OUT_DIM, o = i % OUT_DIM;
    out[i] = outT[o * B_DIM + b];
}

// ---------------- launch ----------------
extern "C" void kernel_launch(void* const* d_in, const int* in_sizes, int n_in,
                              void* d_out, int out_size, void* d_ws, size_t ws_size,
                              hipStream_t stream) {
    const float* x     = (const float*)d_in[0];
    const float* z     = (const float*)d_in[1];
    const float* Wnet  = (const float*)d_in[2];
    const float* bnet  = (const float*)d_in[3];
    const int*   inidx = (const int*)d_in[4];
    const int*   outix = (const int*)d_in[5];
    float* out = (float*)d_out;

    float* xT   = (float*)d_ws;                     // [IN_DIM  x 32]
    float* outT = xT + (size_t)IN_DIM * B_DIM;      // [OUT_DIM x 32]

    hyper_pre <<<(B_DIM * IN_DIM) / 256, 256, 0, stream>>>(x, xT, outT);
    hyper_fused<<<NTILES, 256, 0, stream>>>(xT, z, Wnet, bnet, inidx, outix, outT);
    hyper_post<<<(B_DIM * OUT_DIM) / 256, 256, 0, stream>>>(outT, out);
}